// ProjectLayer_26508538151747
// MI455X (gfx1250) — compile-verified
//
#include <hip/hip_runtime.h>
#include <math.h>

// ---- config constants (from the reference) ----
#define N_VIEWS   5
#define N_JOINTS  15
#define HM_H      128
#define HM_W      240
#define VX        64                     // cube voxels per axis
#define FINE_XY   253                    // fine grid x,y
#define FINE_Z    64                     // fine grid z
#define CUBE_N    (VX * VX * VX)         // 262144 voxels per cube
#define N_PEOPLE  10
#define HM_JSZ    (HM_H * HM_W)          // 30720 floats per (view,joint) heatmap
#define HM_VSZ    (N_JOINTS * HM_JSZ)    // 460800 floats per view
#define OUT_CUBES ((size_t)N_PEOPLE * N_JOINTS * CUBE_N)   // 39,321,600 floats

// SCALE = (FINE-1).f32 / (CAP_V-1) computed in f64 then cast f32 (matches numpy)
#define SCALE_XY  ((float)(252.0 / 79.0))
#define SCALE_Z   ((float)(63.0 / 19.0))
#define BIAS_ALL  (-31.5f)               // exact in f32 for all three axes

// -------------------------------------------------------------------------
// offset[p][a] = round(pc*scale+bias) / (fine-1) * whole - whole/2 + ind/2
// -------------------------------------------------------------------------
__global__ void project_offset_kernel(const float* __restrict__ pc,
                                      float* __restrict__ out_off) {
  int i = threadIdx.x;
  if (i >= N_PEOPLE * 3) return;
  int p = i / 3, a = i % 3;
  float scale = (a == 2) ? SCALE_Z : SCALE_XY;
  float c = rintf(pc[p * 7 + a] * scale + BIAS_ALL);   // centers_tl as float
  float fine1 = (a == 2) ? 63.0f : 252.0f;
  float whole = (a == 2) ? 2000.0f : 8000.0f;
  out_off[i] = c / fine1 * whole - whole * 0.5f + 1000.0f;  // ind/2 == 1000 all axes
}

// -------------------------------------------------------------------------
// Main kernel: one thread per voxel (z innermost -> coalesced grid loads and
// output stores).
// Phase 1: all 5 views' bilinear weights + clamped tap offsets into registers
//          (decouples grid-coord waits from tap-load waits; LOADcnt is
//          in-order on CDNA5, so mixing them serializes per view).
// Phase 2: per joint, issue the 20 tap loads and FMA-reduce, then NT-store
//          (output is write-once; keep the 192MB L2 for the gather reads).
// -------------------------------------------------------------------------
__global__ __launch_bounds__(256) void project_cube_kernel(
    const float* __restrict__ hm,     // [5,15,128,240]
    const float* __restrict__ grid,   // [5,253,253,64,2]
    const float* __restrict__ pc,     // [10,7]
    float* __restrict__ out)          // [10,15,64,64,64]
{
  const int p = blockIdx.y;
  const int n = blockIdx.x * 256 + threadIdx.x;       // voxel linear id [0, 64^3)

  // Warm L2 with the (small, fully cacheable) heatmap working set.
  // n*8 spans ~8.4MB of the 9.2MB heatmap buffer.  (global_prefetch_b8)
  __builtin_prefetch(hm + ((size_t)n << 3), 0, 0);

  // ---- per-person scalar setup (uniform across block; cheap) ----
  const float px = pc[p * 7 + 0], py = pc[p * 7 + 1], pz = pc[p * 7 + 2];
  const float sx = pc[p * 7 + 5], sy = pc[p * 7 + 6];

  const int cx = (int)rintf(px * SCALE_XY + BIAS_ALL);
  const int cy = (int)rintf(py * SCALE_XY + BIAS_ALL);
  const int cz = (int)rintf(pz * SCALE_Z  + BIAS_ALL);

  int mx = (int)((1.0f - sx) * 0.5f * 63.0f); if (mx < 0) mx = 0;   // trunc = astype
  int my = (int)((1.0f - sy) * 0.5f * 63.0f); if (my < 0) my = 0;

  const int stx = max(cx + mx, 0), sty = max(cy + my, 0), stz = max(cz, 0);
  const int enx = min(cx + VX - mx, FINE_XY);
  const int eny = min(cy + VX - my, FINE_XY);
  const int enz = min(cz + VX, FINE_Z);
  const bool skip = (stx >= enx) || (sty >= eny) || (stz >= enz);

  // ---- voxel coordinates (dz innermost) ----
  const int dz = n & 63, dy = (n >> 6) & 63, dx = n >> 12;
  const int fx = cx + dx, fy = cy + dy, fz = cz + dz;
  const bool valid = !skip &&
                     (fx >= stx) && (fx < enx) &&
                     (fy >= sty) && (fy < eny) &&
                     (fz >= stz) && (fz < enz);
  const int fxc = min(max(fx, 0), FINE_XY - 1);
  const int fyc = min(max(fy, 0), FINE_XY - 1);
  const int fzc = min(max(fz, 0), FINE_Z - 1);
  const int gbase = (((fxc * FINE_XY) + fyc) * FINE_Z + fzc) * 2;

  // ---- phase 1: per-view bilinear state, fully register-resident ----
  float w[N_VIEWS][4];   // corner weights with zeros-padding masks folded in
  int   o[N_VIEWS][4];   // tap offsets incl. view base (add j*HM_JSZ per joint)

#pragma unroll
  for (int v = 0; v < N_VIEWS; ++v) {
    // coalesced 8B-aligned load of (gx, gy)
    const float2 g =
        *(const float2*)(grid + (size_t)v * (FINE_XY * FINE_XY * FINE_Z * 2) + gbase);

    const float x = (g.x + 1.0f) * 0.5f * (float)(HM_W - 1);
    const float y = (g.y + 1.0f) * 0.5f * (float)(HM_H - 1);
    const float xf = floorf(x), yf = floorf(y);
    const int   x0 = (int)xf,   y0 = (int)yf;
    const float wx = x - xf,    wy = y - yf;

    const bool xl = (x0 >= 0) && (x0 < HM_W);
    const bool xr = (x0 >= -1) && (x0 < HM_W - 1);
    const bool yt = (y0 >= 0) && (y0 < HM_H);
    const bool yb = (y0 >= -1) && (y0 < HM_H - 1);
    w[v][0] = (xl && yt) ? (1.0f - wx) * (1.0f - wy) : 0.0f;
    w[v][1] = (xr && yt) ? wx * (1.0f - wy)          : 0.0f;
    w[v][2] = (xl && yb) ? (1.0f - wx) * wy          : 0.0f;
    w[v][3] = (xr && yb) ? wx * wy                   : 0.0f;

    // clamped tap addresses (always in-bounds; masked weights kill OOB taps)
    const int x0c = min(max(x0, 0),     HM_W - 1);
    const int x1c = min(max(x0 + 1, 0), HM_W - 1);
    const int y0c = min(max(y0, 0),     HM_H - 1);
    const int y1c = min(max(y0 + 1, 0), HM_H - 1);
    const int vb  = v * HM_VSZ;
    o[v][0] = vb + y0c * HM_W + x0c;
    o[v][1] = vb + y0c * HM_W + x1c;
    o[v][2] = vb + y1c * HM_W + x0c;
    o[v][3] = vb + y1c * HM_W + x1c;
  }

  // ---- phase 2: per joint, 20 gathers -> FMA reduce -> NT streaming store ----
  const float vm = valid ? 1.0f : 0.0f;
  const size_t ob = (size_t)p * N_JOINTS * CUBE_N + (size_t)n;

#pragma unroll
  for (int j = 0; j < N_JOINTS; ++j) {
    const int jb = j * HM_JSZ;
    float t = 0.0f;
#pragma unroll
    for (int v = 0; v < N_VIEWS; ++v) {
      const float s0 = hm[o[v][0] + jb];
      const float s1 = hm[o[v][1] + jb];
      const float s2 = hm[o[v][2] + jb];
      const float s3 = hm[o[v][3] + jb];
      float sv = w[v][0] * s0;
      sv = fmaf(w[v][1], s1, sv);
      sv = fmaf(w[v][2], s2, sv);
      sv = fmaf(w[v][3], s3, sv);
      t += sv;
    }
    float c = t * 0.2f;                      // mean over 5 views
    c = fminf(fmaxf(c, 0.0f), 1.0f);         // clip [0,1]
    __builtin_nontemporal_store(c * vm, &out[ob + (size_t)j * CUBE_N]);
  }
}

extern "C" void kernel_launch(void* const* d_in, const int* in_sizes, int n_in,
                              void* d_out, int out_size, void* d_ws, size_t ws_size,
                              hipStream_t stream) {
  (void)in_sizes; (void)n_in; (void)d_ws; (void)ws_size; (void)out_size;
  const float* hm   = (const float*)d_in[0];   // [5,15,128,240]
  const float* grid = (const float*)d_in[1];   // [5,253,253,64,2]
  const float* pc   = (const float*)d_in[2];   // [10,7]
  float* out = (float*)d_out;

  // cubes: 10*15*64^3 floats, then offset: 10*3 floats
  dim3 grid_dim(CUBE_N / 256, N_PEOPLE);
  project_cube_kernel<<<grid_dim, 256, 0, stream>>>(hm, grid, pc, out);
  project_offset_kernel<<<1, 32, 0, stream>>>(pc, out + OUT_CUBES);
}